// Spca_layer_46866683134189
// MI455X (gfx1250) — compile-verified
//
#include <hip/hip_runtime.h>
#include <hip/hip_bf16.h>

#define B_ 16
#define CIN 256
#define H_ 48
#define W_ 48
#define N_ (H_ * W_)      // 2304
#define K_ 64
#define CD_ 64
#define GAMMA_F ((float)(1.0 - 1e-10))

typedef __attribute__((ext_vector_type(2))) float v2f;
typedef __attribute__((ext_vector_type(8))) float v8f;
typedef __attribute__((ext_vector_type(4))) int   v4i;

#if __has_builtin(__builtin_amdgcn_global_load_async_to_lds_b128)
#define HAVE_ASYNC_LDS 1
#endif

// Async global->LDS 16B copy (ASYNCcnt-tracked); sync fallback keeps the
// pipeline structure valid (dscnt covered by __syncthreads).
// Builtin prototype (per hipcc diagnostic): first arg is non-const
// addrspace(1) pointer to int4; LDS arg mirrors it in addrspace(3).
__device__ __forceinline__ void async_ld_b128(const float* g, float* l)
{
#ifdef HAVE_ASYNC_LDS
    v4i* gg = (v4i*)(float*)g;   // drop const, reinterpret (generic)
    v4i* ll = (v4i*)l;
    __builtin_amdgcn_global_load_async_to_lds_b128(
        (__attribute__((address_space(1))) v4i*)gg,
        (__attribute__((address_space(3))) v4i*)ll, 0, 0);
#else
    *(float4*)l = *(const float4*)g;
#endif
}

__device__ __forceinline__ void wait_async_0()
{
#ifdef HAVE_ASYNC_LDS
#if __has_builtin(__builtin_amdgcn_s_wait_asynccnt)
    __builtin_amdgcn_s_wait_asynccnt(0);
#else
    asm volatile("s_wait_asynccnt 0x0" ::: "memory");
#endif
#endif
}

__device__ __forceinline__ void wait_async_2()
{
#ifdef HAVE_ASYNC_LDS
#if __has_builtin(__builtin_amdgcn_s_wait_asynccnt)
    __builtin_amdgcn_s_wait_asynccnt(2);
#else
    asm volatile("s_wait_asynccnt 0x2" ::: "memory");
#endif
#endif
}

// ---------------------------------------------------------------------------
// Generic fp32 WMMA GEMM: C[M,N] = A[M,K] * B[K,N], row-major, optional C^T
// store. Block = 128 threads (4 waves), block tile 64x64, K-step 16,
// double-buffered LDS. A tile staged with async-to-LDS b128; B tile software-
// pipelined through registers and stored COLUMN-MAJOR in LDS so both WMMA
// fragments are single aligned ds_load_b64 (no v_mov repacking).
// Requires M%64==0 (or M==64, gridDim.y==1), N%64==0, K%16==0, lds %4==0.
// ---------------------------------------------------------------------------
__launch_bounds__(128)
__global__ void gemm_f32_wmma(const float* __restrict__ A,
                              const float* __restrict__ B,
                              float* __restrict__ C,
                              int M, int N, int K,
                              int lda, int ldb, int ldc,
                              long sA, long sB, long sC, int transC)
{
    __shared__ float As[2][64][20];   // [buf][m][k]  row = 80B (16B aligned)
    __shared__ float Bs[2][64][18];   // [buf][n][k]  row = 72B (8B aligned, ko even)

    const int bz = blockIdx.z;
    A += (long)bz * sA;
    B += (long)bz * sB;
    C += (long)bz * sC;

    const int tileM = blockIdx.y * 64;
    const int tileN = blockIdx.x * 64;
    const int tid   = threadIdx.x;
    const int wv    = tid >> 5;
    const int lane  = tid & 31;
    const int lhalf = lane >> 4;
    const int l16   = lane & 15;

    // staging slot decomposition (2 x 16B slots per thread for each tile)
    const int ar0 = tid >> 2,          aq0 = (tid & 3) * 4;
    const int ar1 = (tid + 128) >> 2,  aq1 = aq0;            // idx+128 keeps q
    const int bk0 = tid >> 4,          bn0 = (tid & 15) * 4;
    const int bk1 = (tid + 128) >> 4,  bn1 = bn0;

    v8f acc0 = {}, acc1 = {}, acc2 = {}, acc3 = {};

    const int T = K >> 4;

    // -------- prologue: tile 0 --------
    async_ld_b128(A + (long)(tileM + ar0) * lda + aq0, &As[0][ar0][aq0]);
    async_ld_b128(A + (long)(tileM + ar1) * lda + aq1, &As[0][ar1][aq1]);
    float4 breg0 = *(const float4*)(B + (long)bk0 * ldb + tileN + bn0);
    float4 breg1 = *(const float4*)(B + (long)bk1 * ldb + tileN + bn1);

    for (int t = 0; t < T; ++t) {
        const int p  = t & 1;
        const int kt = t << 4;

        // publish B regs (tile t) column-major into Bs[p]
        Bs[p][bn0 + 0][bk0] = breg0.x;
        Bs[p][bn0 + 1][bk0] = breg0.y;
        Bs[p][bn0 + 2][bk0] = breg0.z;
        Bs[p][bn0 + 3][bk0] = breg0.w;
        Bs[p][bn1 + 0][bk1] = breg1.x;
        Bs[p][bn1 + 1][bk1] = breg1.y;
        Bs[p][bn1 + 2][bk1] = breg1.z;
        Bs[p][bn1 + 3][bk1] = breg1.w;

        const bool has_next = (t + 1 < T);
        if (has_next) {
            const int kn = kt + 16;
            // prefetch A tile t+1 into the other buffer (async, in flight
            // through the whole compute phase below)
            async_ld_b128(A + (long)(tileM + ar0) * lda + kn + aq0, &As[p ^ 1][ar0][aq0]);
            async_ld_b128(A + (long)(tileM + ar1) * lda + kn + aq1, &As[p ^ 1][ar1][aq1]);
            // prefetch B tile t+1 into registers
            breg0 = *(const float4*)(B + (long)(kn + bk0) * ldb + tileN + bn0);
            breg1 = *(const float4*)(B + (long)(kn + bk1) * ldb + tileN + bn1);
            wait_async_2();   // tile t's 2 async ops done; t+1's may remain
        } else {
            wait_async_0();
        }
        __syncthreads();      // all waves' A(t) + B(t) visible in LDS

#pragma unroll
        for (int ks = 0; ks < 16; ks += 4) {
            const int ko = ks + 2 * lhalf;
            v2f a  = *(const v2f*)&As[p][wv * 16 + l16][ko];
            v2f b0 = *(const v2f*)&Bs[p][l16][ko];
            v2f b1 = *(const v2f*)&Bs[p][16 + l16][ko];
            v2f b2 = *(const v2f*)&Bs[p][32 + l16][ko];
            v2f b3 = *(const v2f*)&Bs[p][48 + l16][ko];
            acc0 = __builtin_amdgcn_wmma_f32_16x16x4_f32(false, a, false, b0, (short)0, acc0, false, false);
            acc1 = __builtin_amdgcn_wmma_f32_16x16x4_f32(false, a, false, b1, (short)0, acc1, false, false);
            acc2 = __builtin_amdgcn_wmma_f32_16x16x4_f32(false, a, false, b2, (short)0, acc2, false, false);
            acc3 = __builtin_amdgcn_wmma_f32_16x16x4_f32(false, a, false, b3, (short)0, acc3, false, false);
        }
        __syncthreads();      // buffers free for the next iteration's writes
    }

    // D layout: VGPR r -> M = r + 8*lhalf ; N = l16
    const int mbase = tileM + wv * 16 + lhalf * 8;
#pragma unroll
    for (int r = 0; r < 8; ++r) {
        const int mm = mbase + r;
        float v0 = acc0[r], v1 = acc1[r], v2 = acc2[r], v3 = acc3[r];
        if (!transC) {
            float* cr = C + (long)mm * ldc + tileN + l16;
            cr[0] = v0; cr[16] = v1; cr[32] = v2; cr[48] = v3;
        } else {
            C[(long)(tileN +      l16) * ldc + mm] = v0;
            C[(long)(tileN + 16 + l16) * ldc + mm] = v1;
            C[(long)(tileN + 32 + l16) * ldc + mm] = v2;
            C[(long)(tileN + 48 + l16) * ldc + mm] = v3;
        }
    }
}

// ---------------------------------------------------------------------------
// LayerNorm of clusters[K_,CD_] -> tT[CD_][K_] and t2[k] = ||t_k||^2.
// ---------------------------------------------------------------------------
__launch_bounds__(64)
__global__ void ln_templates(const float* __restrict__ clusters,
                             const float* __restrict__ scale,
                             const float* __restrict__ bias,
                             float* __restrict__ tT, float* __restrict__ t2)
{
    const int k = blockIdx.x, c = threadIdx.x;
    __shared__ float red[64];
    float v = clusters[k * CD_ + c];
    red[c] = v; __syncthreads();
    for (int s = 32; s > 0; s >>= 1) { if (c < s) red[c] += red[c + s]; __syncthreads(); }
    float mean = red[0] * (1.0f / CD_); __syncthreads();
    float d = v - mean;
    red[c] = d * d; __syncthreads();
    for (int s = 32; s > 0; s >>= 1) { if (c < s) red[c] += red[c + s]; __syncthreads(); }
    float var = red[0] * (1.0f / CD_); __syncthreads();
    float tn = d * rsqrtf(var + 1e-5f) * scale[c] + bias[c];
    tT[c * K_ + k] = tn;
    red[c] = tn * tn; __syncthreads();
    for (int s = 32; s > 0; s >>= 1) { if (c < s) red[c] += red[c + s]; __syncthreads(); }
    if (c == 0) t2[k] = red[0];
}

// ---------------------------------------------------------------------------
// Per (b,n): logits from P = xf@tT, softmax over K; write attn (transposed),
// raw attn_n (transposed), atomic column sums, x2. One wave32 per (b,n).
// ---------------------------------------------------------------------------
__launch_bounds__(256)
__global__ void softmax_kernel(const float* __restrict__ xf,
                               const float* __restrict__ P,
                               const float* __restrict__ t2,
                               const float* __restrict__ pi,
                               const float* __restrict__ cov,
                               float* __restrict__ attn_out,
                               float* __restrict__ raw,
                               float* __restrict__ colsum,
                               float* __restrict__ denom,
                               float* __restrict__ x2buf)
{
    const int wid  = blockIdx.x * 8 + (threadIdx.x >> 5);
    const int lane = threadIdx.x & 31;
    const int b = wid / N_, n = wid % N_;

    const float* xr = xf + ((long)b * N_ + n) * CD_;
    float v0 = xr[lane], v1 = xr[lane + 32];
    float x2 = v0 * v0 + v1 * v1;
    for (int o = 16; o; o >>= 1) x2 += __shfl_down(x2, o, 32);
    x2 = __shfl(x2, 0, 32);

    const float* pr = P + ((long)b * N_ + n) * K_;
    const int k0 = lane, k1 = lane + 32;
    float c0 = cov[k0], c1 = cov[k1];
    float l0 = logf(pi[k0] * rsqrtf(c0)) - (x2 + t2[k0] - 2.0f * pr[k0]) * (0.5f / c0);
    float l1 = logf(pi[k1] * rsqrtf(c1)) - (x2 + t2[k1] - 2.0f * pr[k1]) * (0.5f / c1);

    float mx = fmaxf(l0, l1);
    for (int o = 16; o; o >>= 1) mx = fmaxf(mx, __shfl_down(mx, o, 32));
    mx = __shfl(mx, 0, 32);
    float e0 = expf(l0 - mx), e1 = expf(l1 - mx);
    float sm = e0 + e1;
    for (int o = 16; o; o >>= 1) sm += __shfl_down(sm, o, 32);
    sm = __shfl(sm, 0, 32);
    float a0 = e0 / sm, a1 = e1 / sm;

    const long ob = (long)b * K_ * N_;
    attn_out[ob + (long)k0 * N_ + n] = a0;
    attn_out[ob + (long)k1 * N_ + n] = a1;
    float r0 = a0 + 1e-8f, r1 = a1 + 1e-8f;
    raw[ob + (long)k0 * N_ + n] = r0;
    raw[ob + (long)k1 * N_ + n] = r1;
    atomicAdd(&colsum[b * K_ + k0], a0);
    atomicAdd(&colsum[b * K_ + k1], a1);
    atomicAdd(&denom[b * K_ + k0], r0);
    atomicAdd(&denom[b * K_ + k1], r1);
    if (lane == 0) x2buf[b * N_ + n] = x2;
}

// ---------------------------------------------------------------------------
// Per (b,k): attn_nT /= denom; q = sum_n attn_n*x2; pi_new.
// ---------------------------------------------------------------------------
__launch_bounds__(256)
__global__ void normalize_kernel(float* __restrict__ attn_nT,
                                 const float* __restrict__ denom,
                                 const float* __restrict__ x2buf,
                                 const float* __restrict__ colsum,
                                 const float* __restrict__ pi,
                                 float* __restrict__ q,
                                 float* __restrict__ pi_new)
{
    const int bk = blockIdx.x;
    const int b = bk >> 6, k = bk & 63;
    const float inv = 1.0f / denom[bk];
    float* row = attn_nT + (long)bk * N_;
    const float* x2r = x2buf + b * N_;
    float acc = 0.f;
    for (int n = threadIdx.x; n < N_; n += 256) {
        float v = row[n] * inv;
        row[n] = v;
        acc += v * x2r[n];
    }
    __shared__ float red[256];
    red[threadIdx.x] = acc; __syncthreads();
    for (int s = 128; s > 0; s >>= 1) { if (threadIdx.x < s) red[threadIdx.x] += red[threadIdx.x + s]; __syncthreads(); }
    if (threadIdx.x == 0) {
        q[bk] = red[0];
        float p = pi[k];
        pi_new[bk] = p + GAMMA_F * (colsum[bk] * (1.0f / N_) - p);
    }
}

// ---------------------------------------------------------------------------
// t_new = clusters + g*(S - clusters); cov_new via q + ||t_new||^2 - 2<S,t_new>
// (uses sum_n attn_n == 1). One wave per (b,k).
// ---------------------------------------------------------------------------
__launch_bounds__(256)
__global__ void finalize_t(const float* __restrict__ S,
                           const float* __restrict__ clusters,
                           const float* __restrict__ cov,
                           const float* __restrict__ q,
                           float* __restrict__ t_new,
                           float* __restrict__ cov_new)
{
    const int wid  = blockIdx.x * 8 + (threadIdx.x >> 5);
    const int lane = threadIdx.x & 31;
    const int k = wid & 63;
    const float* sr = S + (long)wid * CD_;
    const float* cl = clusters + k * CD_;
    float s0 = sr[lane], s1 = sr[lane + 32];
    float c0 = cl[lane], c1 = cl[lane + 32];
    float t0 = c0 + GAMMA_F * (s0 - c0);
    float t1 = c1 + GAMMA_F * (s1 - c1);
    t_new[(long)wid * CD_ + lane]      = t0;
    t_new[(long)wid * CD_ + lane + 32] = t1;
    float tn2 = t0 * t0 + t1 * t1;
    float dst = s0 * t0 + s1 * t1;
    for (int o = 16; o; o >>= 1) { tn2 += __shfl_down(tn2, o, 32); dst += __shfl_down(dst, o, 32); }
    if (lane == 0) {
        float cv = cov[k];
        cov_new[wid] = cv + GAMMA_F * (q[wid] + tn2 - 2.0f * dst - cv);
    }
}

// ---------------------------------------------------------------------------
// Spatial mask: column sums of exp(-dist) (symmetric), then A, then A@A.
// ---------------------------------------------------------------------------
__launch_bounds__(256)
__global__ void mask_rowsum(float* __restrict__ srow)
{
    const int m = blockIdx.x;
    const float im = (float)(m / W_), jm = (float)(m % W_);
    float acc = 0.f;
    for (int n = threadIdx.x; n < N_; n += 256) {
        float di = (float)(n / W_) - im, dj = (float)(n % W_) - jm;
        acc += expf(-sqrtf(di * di + dj * dj));
    }
    __shared__ float red[256];
    red[threadIdx.x] = acc; __syncthreads();
    for (int s = 128; s > 0; s >>= 1) { if (threadIdx.x < s) red[threadIdx.x] += red[threadIdx.x + s]; __syncthreads(); }
    if (threadIdx.x == 0) srow[m] = red[0];
}

__launch_bounds__(256)
__global__ void build_A(const float* __restrict__ srow, float* __restrict__ Amat)
{
    const long idx = (long)blockIdx.x * 256 + threadIdx.x;
    const int n = (int)(idx / N_), m = (int)(idx % N_);
    float di = (float)(n / W_) - (float)(m / W_);
    float dj = (float)(n % W_) - (float)(m % W_);
    Amat[idx] = expf(-sqrtf(di * di + dj * dj)) / srow[m];
}

// ---------------------------------------------------------------------------
// Final LN over K per (b,m) via LDS transpose; 64 threads own a 64-wide m-tile.
// ---------------------------------------------------------------------------
__launch_bounds__(64)
__global__ void outp_ln(const float* __restrict__ raw,
                        const float* __restrict__ scale,
                        const float* __restrict__ bias,
                        float* __restrict__ outp)
{
    const int blk = blockIdx.x;
    const int b  = blk / (N_ / 64);
    const int m0 = (blk % (N_ / 64)) * 64;
    const int tid = threadIdx.x;
    __shared__ float tile[64][65];
    const float* base = raw + (long)b * K_ * N_;
    for (int k = 0; k < 64; ++k) tile[k][tid] = base[(long)k * N_ + m0 + tid];
    __syncthreads();
    float mean = 0.f;
    for (int k = 0; k < 64; ++k) mean += tile[k][tid];
    mean *= (1.0f / 64.0f);
    float var = 0.f;
    for (int k = 0; k < 64; ++k) { float d = tile[k][tid] - mean; var += d * d; }
    var *= (1.0f / 64.0f);
    float inv = rsqrtf(var + 1e-5f);
    float* ob = outp + (long)b * K_ * N_;
    for (int k = 0; k < 64; ++k)
        ob[(long)k * N_ + m0 + tid] = (tile[k][tid] - mean) * inv * scale[k] + bias[k];
}

// ---------------------------------------------------------------------------
extern "C" void kernel_launch(void* const* d_in, const int* in_sizes, int n_in,
                              void* d_out, int out_size, void* d_ws, size_t ws_size,
                              hipStream_t stream)
{
    const float* x        = (const float*)d_in[0];
    const float* clusters = (const float*)d_in[1];
    const float* pi       = (const float*)d_in[2];
    const float* cov      = (const float*)d_in[3];
    const float* w_proj   = (const float*)d_in[4];
    const float* nt_scale = (const float*)d_in[5];
    const float* nt_bias  = (const float*)d_in[6];
    const float* pn_scale = (const float*)d_in[7];
    const float* pn_bias  = (const float*)d_in[8];

    float* out = (float*)d_out;
    float* outp_out = out;                                   // [B,K,N]
    float* attn_out = outp_out + (long)B_ * K_ * N_;         // [B,K,N]
    float* tnew_out = attn_out + (long)B_ * K_ * N_;         // [B,K,CD]
    float* pi_out   = tnew_out + (long)B_ * K_ * CD_;        // [B,K]
    float* cov_out  = pi_out + B_ * K_;                      // [B,K]

    float* ws = (float*)d_ws;
    long off = 0;
    float* xf      = ws + off; off += (long)B_ * N_ * CD_;   // [B,N,CD]
    float* P       = ws + off; off += (long)B_ * N_ * K_;    // [B,N,K]
    float* attn_nT = ws + off; off += (long)B_ * K_ * N_;    // [B,K,N]
    float* x2buf   = ws + off; off += (long)B_ * N_;
    float* colsum  = ws + off; off += B_ * K_;
    float* denom   = ws + off; off += B_ * K_;
    float* q       = ws + off; off += B_ * K_;
    float* S       = ws + off; off += (long)B_ * K_ * CD_;
    float* tT      = ws + off; off += K_ * CD_;
    float* t2      = ws + off; off += K_;
    float* srow    = ws + off; off += N_;
    float* Amat    = ws + off; off += (long)N_ * N_;
    float* A2      = ws + off; off += (long)N_ * N_;
    float* outp_raw = Amat;   // alias: A dead once A2 = A@A done

    (void)hipMemsetAsync(colsum, 0, (size_t)2 * B_ * K_ * sizeof(float), stream);

    ln_templates<<<K_, 64, 0, stream>>>(clusters, nt_scale, nt_bias, tT, t2);

    // xf[b] = (w_proj @ x[b])^T : M=64, N=2304, K=256, store transposed
    {
        dim3 g(N_ / 64, 1, B_);
        gemm_f32_wmma<<<g, 128, 0, stream>>>(w_proj, x, xf,
                                             64, N_, CIN,
                                             CIN, N_, CD_,
                                             0L, (long)CIN * N_, (long)N_ * CD_, 1);
    }
    // P[b] = xf[b] @ tT : M=2304, N=64, K=64
    {
        dim3 g(1, N_ / 64, B_);
        gemm_f32_wmma<<<g, 128, 0, stream>>>(xf, tT, P,
                                             N_, 64, 64,
                                             CD_, K_, K_,
                                             (long)N_ * CD_, 0L, (long)N_ * K_, 0);
    }
    softmax_kernel<<<B_ * N_ / 8, 256, 0, stream>>>(xf, P, t2, pi, cov,
                                                    attn_out, attn_nT,
                                                    colsum, denom, x2buf);
    normalize_kernel<<<B_ * K_, 256, 0, stream>>>(attn_nT, denom, x2buf,
                                                  colsum, pi, q, pi_out);
    // S[b] = attn_nT[b] @ xf[b] : M=64, N=64, K=2304
    {
        dim3 g(1, 1, B_);
        gemm_f32_wmma<<<g, 128, 0, stream>>>(attn_nT, xf, S,
                                             64, 64, N_,
                                             N_, CD_, CD_,
                                             (long)K_ * N_, (long)N_ * CD_, (long)K_ * CD_, 0);
    }
    finalize_t<<<B_ * K_ / 8, 256, 0, stream>>>(S, clusters, cov, q, tnew_out, cov_out);

    mask_rowsum<<<N_, 256, 0, stream>>>(srow);
    build_A<<<(int)(((long)N_ * N_) / 256), 256, 0, stream>>>(srow, Amat);

    // A2 = A @ A : 2304^3 (dominant GEMM)
    {
        dim3 g(N_ / 64, N_ / 64, 1);
        gemm_f32_wmma<<<g, 128, 0, stream>>>(Amat, Amat, A2,
                                             N_, N_, N_,
                                             N_, N_, N_,
                                             0L, 0L, 0L, 0);
    }
    // outp_raw[b] = attn_nT[b] @ A2 : M=64, N=2304, K=2304
    {
        dim3 g(N_ / 64, 1, B_);
        gemm_f32_wmma<<<g, 128, 0, stream>>>(attn_nT, A2, outp_raw,
                                             64, N_, N_,
                                             N_, N_, N_,
                                             (long)K_ * N_, 0L, (long)K_ * N_, 0);
    }
    outp_ln<<<B_ * (N_ / 64), 64, 0, stream>>>(outp_raw, pn_scale, pn_bias, outp_out);
}